// Retrieve_MRR_15573551415404
// MI455X (gfx1250) — compile-verified
//
#include <hip/hip_runtime.h>

// Retrieve-MRR for N=8192, D=1024, K=4 on MI455X (gfx1250).
//
// Key identity: for a stable descending argsort of continuous data,
//   min_rank(i) = #{ j : sim[i,j] > max_k sim[i, gt[i,k]] }
// so we never materialize or sort the 8192x8192 sim matrix:
//   A) per-row threshold t[i] = max_k dot(m1[i], m2[gt[i,k]])   (~67 MFLOP)
//   B) fp32 WMMA GEMM (137 GFLOP) whose epilogue *counts* sim > t[i]
//   C) mean(1/(count+1))
//
// Roofline: both matrices (32 MB each) are L2-resident (192 MB L2); with a
// 128x128 block tile the L2 read traffic is N^2*D*4*(2/128) ~= 4.3 GB, so the
// kernel is WMMA-compute-bound. fp32 (V_WMMA_F32_16X16X4_F32) is mandatory:
// the count-vs-threshold comparison is tie-sensitive.
//
// CDNA5 data movement: global->LDS staging uses GLOBAL_LOAD_ASYNC_TO_LDS_B128
// (ASYNCcnt-tracked DMA, no VGPR round-trip) with LDS double buffering so the
// async copies of chunk c+1 overlap the 64 WMMAs of chunk c.

#define NROWS 8192
#define DDIM  1024
#define KGT   4

#define BM 128
#define BN 128
#define BK 32
#define NCHUNK (DDIM / BK)
#define LDS_STRIDE 36   // even (float2 fragment reads 8B-aligned), 36*4=144 B
                        // (16B-aligned b128 stores), and 36*r mod 64 distinct
                        // for r=0..15 -> conflict-free 16-row fragment reads.

typedef __attribute__((ext_vector_type(2))) float v2f;
typedef __attribute__((ext_vector_type(8))) float v8f;

// Pointee type the async-LDS builtin expects (per hipcc diagnostic):
// "__attribute__((__vector_size__(4 * sizeof(int)))) int *"
typedef int v4i_t __attribute__((vector_size(4 * sizeof(int))));

// ---------------------------------------------------------------------------
// CDNA5 async global->LDS copy helpers
// ---------------------------------------------------------------------------
__device__ __forceinline__ void async_copy_b128(const float* g, float* l) {
#if __has_builtin(__builtin_amdgcn_global_load_async_to_lds_b128)
  __builtin_amdgcn_global_load_async_to_lds_b128(
      (__attribute__((address_space(1))) v4i_t*)(g),
      (__attribute__((address_space(3))) v4i_t*)(l),
      /*offset=*/0, /*cpol=*/0);
#else
  const unsigned loff =
      (unsigned)(unsigned long long)(__attribute__((address_space(3))) void*)(l);
  asm volatile("global_load_async_to_lds_b128 %0, %1, off"
               :
               : "v"(loff), "v"(g)
               : "memory");
#endif
}

__device__ __forceinline__ void wait_async0() {
#if __has_builtin(__builtin_amdgcn_s_wait_asynccnt)
  __builtin_amdgcn_s_wait_asynccnt(0);
#else
  asm volatile("s_wait_asynccnt 0x0" ::: "memory");
#endif
}

// ---------------------------------------------------------------------------
// Phase A: t[i] = max_k dot(m1[i], m2[gt[i,k]])   (one wave32 per row)
// ---------------------------------------------------------------------------
__global__ __launch_bounds__(256)
void mrr_thresh_kernel(const float* __restrict__ m1,
                       const float* __restrict__ m2,
                       const int*   __restrict__ gt,
                       float*       __restrict__ thresh) {
  const int wave = threadIdx.x >> 5;
  const int lane = threadIdx.x & 31;
  const int row  = blockIdx.x * 8 + wave;

  const int g0 = gt[row * KGT + 0];
  const int g1 = gt[row * KGT + 1];
  const int g2 = gt[row * KGT + 2];
  const int g3 = gt[row * KGT + 3];

  const float* a  = m1 + (size_t)row * DDIM;
  const float* b0 = m2 + (size_t)g0 * DDIM;
  const float* b1 = m2 + (size_t)g1 * DDIM;
  const float* b2 = m2 + (size_t)g2 * DDIM;
  const float* b3 = m2 + (size_t)g3 * DDIM;

  float s0 = 0.f, s1 = 0.f, s2 = 0.f, s3 = 0.f;
  for (int d = lane; d < DDIM; d += 32) {
    const float av = a[d];
    s0 += av * b0[d];
    s1 += av * b1[d];
    s2 += av * b2[d];
    s3 += av * b3[d];
  }
  #pragma unroll
  for (int off = 16; off > 0; off >>= 1) {
    s0 += __shfl_down(s0, off, 32);
    s1 += __shfl_down(s1, off, 32);
    s2 += __shfl_down(s2, off, 32);
    s3 += __shfl_down(s3, off, 32);
  }
  if (lane == 0) {
    thresh[row] = fmaxf(fmaxf(s0, s1), fmaxf(s2, s3));
  }
}

// ---------------------------------------------------------------------------
// Phase B: WMMA fp32 GEMM with counting epilogue.
// 256 threads = 8 waves; block tile 128x128; wave tile 32x64 (2x4 v8f accs).
// A/B chunks (128x32 fp32) double-buffered in LDS via async DMA.
// ---------------------------------------------------------------------------
__global__ __launch_bounds__(256)
void mrr_count_kernel(const float* __restrict__ m1,
                      const float* __restrict__ m2,
                      const float* __restrict__ thresh,
                      int*         __restrict__ rank) {
  __shared__ float ldsA[2][BM * LDS_STRIDE];
  __shared__ float ldsB[2][BN * LDS_STRIDE];

  const int tid  = threadIdx.x;
  const int wave = tid >> 5;       // 0..7
  const int lane = tid & 31;
  const int half = lane >> 4;      // 0: lanes 0-15, 1: lanes 16-31
  const int r    = lane & 15;
  const int wm   = wave >> 1;      // 0..3  (32-row strips)
  const int wn   = wave & 1;       // 0..1  (64-col strips)

  const int mBase = blockIdx.y * BM;
  const int nBase = blockIdx.x * BN;

  // Per-thread staging slot: thread covers rows {tid/8 + 32*i}, segment tid%8.
  const int srow = tid >> 3;       // 0..31
  const int sseg = tid & 7;        // 0..7 (float4 segment within 32-col chunk)

  v8f acc[2][4];
  #pragma unroll
  for (int i = 0; i < 2; ++i)
    #pragma unroll
    for (int q = 0; q < 4; ++q)
      acc[i][q] = (v8f){0.f, 0.f, 0.f, 0.f, 0.f, 0.f, 0.f, 0.f};

  // ---- prologue: async-stage chunk 0 into buffer 0 ----
  #pragma unroll
  for (int i = 0; i < 4; ++i) {
    const int row = srow + 32 * i;
    async_copy_b128(m1 + (size_t)(mBase + row) * DDIM + sseg * 4,
                    &ldsA[0][row * LDS_STRIDE + sseg * 4]);
    async_copy_b128(m2 + (size_t)(nBase + row) * DDIM + sseg * 4,
                    &ldsB[0][row * LDS_STRIDE + sseg * 4]);
  }
  wait_async0();
  __syncthreads();

  #pragma unroll 1
  for (int c = 0; c < NCHUNK; ++c) {
    const int cur = c & 1;

    // ---- async-prefetch chunk c+1 into the other buffer ----
    if (c + 1 < NCHUNK) {
      const int kb = (c + 1) * BK;
      #pragma unroll
      for (int i = 0; i < 4; ++i) {
        const int row = srow + 32 * i;
        async_copy_b128(m1 + (size_t)(mBase + row) * DDIM + kb + sseg * 4,
                        &ldsA[cur ^ 1][row * LDS_STRIDE + sseg * 4]);
        async_copy_b128(m2 + (size_t)(nBase + row) * DDIM + kb + sseg * 4,
                        &ldsB[cur ^ 1][row * LDS_STRIDE + sseg * 4]);
      }
    }

    // ---- compute on chunk c (64 WMMAs) ----
    #pragma unroll
    for (int kk = 0; kk < BK / 4; ++kk) {
      // fp32 A 16x4 fragment layout: lane r holds A[M=r, K=2*half+{0,1}]
      const int kcol = kk * 4 + half * 2;
      v2f afrag[2], bfrag[4];
      #pragma unroll
      for (int i = 0; i < 2; ++i) {
        const float2 t = *(const float2*)(
            &ldsA[cur][(wm * 32 + 16 * i + r) * LDS_STRIDE + kcol]);
        afrag[i].x = t.x;
        afrag[i].y = t.y;
      }
      #pragma unroll
      for (int q = 0; q < 4; ++q) {
        const float2 t = *(const float2*)(
            &ldsB[cur][(wn * 64 + 16 * q + r) * LDS_STRIDE + kcol]);
        bfrag[q].x = t.x;
        bfrag[q].y = t.y;
      }
      #pragma unroll
      for (int i = 0; i < 2; ++i)
        #pragma unroll
        for (int q = 0; q < 4; ++q)
          acc[i][q] = __builtin_amdgcn_wmma_f32_16x16x4_f32(
              false, afrag[i], false, bfrag[q],
              (short)0, acc[i][q], false, false);
    }

    // Reads of ldsX[cur] are complete here (waited before the WMMAs);
    // wait for the prefetch DMA, then barrier before the buffer swap.
    wait_async0();
    __syncthreads();
  }

  // Epilogue: count sim > thresh per row. C/D layout: VGPR j holds row
  // (j + 8*half) of the 16x16 tile, N = lane%16.
  #pragma unroll
  for (int i = 0; i < 2; ++i) {
    #pragma unroll
    for (int j = 0; j < 8; ++j) {
      const int rowLo = mBase + wm * 32 + 16 * i + j;   // half 0 rows
      const int rowHi = rowLo + 8;                      // half 1 rows
      const float tLo = thresh[rowLo];
      const float tHi = thresh[rowHi];
      const float tMine = half ? tHi : tLo;
      #pragma unroll
      for (int q = 0; q < 4; ++q) {
        const bool pred = acc[i][q][j] > tMine;
        const unsigned mask = (unsigned)__ballot(pred);
        if (lane == 0)  atomicAdd(&rank[rowLo], __popc(mask & 0xFFFFu));
        if (lane == 16) atomicAdd(&rank[rowHi], __popc(mask >> 16));
      }
    }
  }
}

// ---------------------------------------------------------------------------
// Phase C: out = mean(1 / (rank + 1))
// ---------------------------------------------------------------------------
__global__ __launch_bounds__(256)
void mrr_finalize_kernel(const int* __restrict__ rank, float* __restrict__ out) {
  __shared__ float red[256];
  float s = 0.f;
  for (int i = threadIdx.x; i < NROWS; i += 256)
    s += 1.0f / (float)(rank[i] + 1);
  red[threadIdx.x] = s;
  __syncthreads();
  #pragma unroll
  for (int off = 128; off > 0; off >>= 1) {
    if (threadIdx.x < off) red[threadIdx.x] += red[threadIdx.x + off];
    __syncthreads();
  }
  if (threadIdx.x == 0) out[0] = red[0] / (float)NROWS;
}

// ---------------------------------------------------------------------------
extern "C" void kernel_launch(void* const* d_in, const int* in_sizes, int n_in,
                              void* d_out, int out_size, void* d_ws, size_t ws_size,
                              hipStream_t stream) {
  (void)in_sizes; (void)n_in; (void)out_size; (void)ws_size;

  const float* m1 = (const float*)d_in[0];
  const float* m2 = (const float*)d_in[1];
  const int*   gt = (const int*)d_in[2];
  float* out = (float*)d_out;

  float* thresh = (float*)d_ws;
  int*   rank   = (int*)((char*)d_ws + NROWS * sizeof(float));

  (void)hipMemsetAsync(rank, 0, NROWS * sizeof(int), stream);

  mrr_thresh_kernel<<<NROWS / 8, 256, 0, stream>>>(m1, m2, gt, thresh);

  dim3 grid(NROWS / BN, NROWS / BM);
  mrr_count_kernel<<<grid, 256, 0, stream>>>(m1, m2, thresh, rank);

  mrr_finalize_kernel<<<1, 256, 0, stream>>>(rank, out);
}